// ApplyAttentionMemory_91001767067663
// MI455X (gfx1250) — compile-verified
//
#include <hip/hip_runtime.h>
#include <hip/hip_bf16.h>
#include <math.h>

typedef float v2f __attribute__((ext_vector_type(2)));
typedef float v8f __attribute__((ext_vector_type(8)));

#define B_ 32
#define L_ 2048
#define E_ 1024
#define F32MAX_ 3.402823466e+38f
#define F32MIN_ (-3.402823466e+38f)

// -------------------------------------------------------------------------
// Kernel 1: scores[b, l] = sum_e query[b,e] * memory[b,l,e]
// One wave computes 16 consecutive l-rows via V_WMMA_F32_16X16X4_F32,
// accumulating over E in K=4 steps.  B operand = query chunk broadcast to
// all 16 columns, so every column of D carries the score.
// A 16x4 f32 layout: lanes 0-15 hold M=lane {K=0(v0),K=1(v1)},
//                    lanes 16-31 hold M=lane-16 {K=2(v0),K=3(v1)}.
// -------------------------------------------------------------------------
__global__ __launch_bounds__(256) void scores_kernel(
    const float* __restrict__ mem, const float* __restrict__ query,
    float* __restrict__ scores) {
  const int lane = threadIdx.x & 31;
  const int wave = threadIdx.x >> 5;
  const int b    = blockIdx.y;
  const int l0   = blockIdx.x * 128 + wave * 16;
  const int m    = lane & 15;
  const int half = lane >> 4;

  const float* __restrict__ arow =
      mem + (size_t)b * L_ * E_ + (size_t)(l0 + m) * E_ + 2 * half;
  const float* __restrict__ q = query + (size_t)b * E_ + 2 * half;

  v8f acc = {0.f, 0.f, 0.f, 0.f, 0.f, 0.f, 0.f, 0.f};
#pragma unroll 4
  for (int e = 0; e < E_; e += 4) {
    v2f a  = *(const v2f*)(arow + e);
    v2f bb = *(const v2f*)(q + e);
    acc = __builtin_amdgcn_wmma_f32_16x16x4_f32(
        false, a, false, bb, (short)0, acc, false, false);
  }
  // C/D layout: VGPR r holds row M=r (lanes 0-15) and M=8+r (lanes 16-31);
  // all N columns identical here.  Lanes 0 and 16 write 8 rows each.
  if (m == 0) {
    float* out = scores + (size_t)b * L_ + l0 + half * 8;
#pragma unroll
    for (int i = 0; i < 8; ++i) out[i] = acc[i];
  }
}

// -------------------------------------------------------------------------
// Kernel 2: masked softmax over L, in-place in the attention slab of d_out.
// One 256-thread block per batch row; 8 elements per thread.
// -------------------------------------------------------------------------
__global__ __launch_bounds__(256) void softmax_kernel(
    float* __restrict__ attn, const int* __restrict__ mask) {
  __shared__ float sred[8];
  const int b   = blockIdx.x;
  const int tid = threadIdx.x;
  const int msk = mask[b];

  float v[8];
  float mx = F32MIN_;
#pragma unroll
  for (int i = 0; i < 8; ++i) {
    const int l = tid + i * 256;
    float s = attn[(size_t)b * L_ + l];
    s = fminf(s, (l < msk) ? F32MAX_ : F32MIN_);
    v[i] = s;
    mx = fmaxf(mx, s);
  }
#pragma unroll
  for (int off = 16; off > 0; off >>= 1) mx = fmaxf(mx, __shfl_xor(mx, off, 32));
  if ((tid & 31) == 0) sred[tid >> 5] = mx;
  __syncthreads();
  float bm = sred[0];
#pragma unroll
  for (int i = 1; i < 8; ++i) bm = fmaxf(bm, sred[i]);
  __syncthreads();

  float sum = 0.f;
#pragma unroll
  for (int i = 0; i < 8; ++i) {
    v[i] = expf(v[i] - bm);
    sum += v[i];
  }
#pragma unroll
  for (int off = 16; off > 0; off >>= 1) sum += __shfl_xor(sum, off, 32);
  if ((tid & 31) == 0) sred[tid >> 5] = sum;
  __syncthreads();
  float tot = sred[0];
#pragma unroll
  for (int i = 1; i < 8; ++i) tot += sred[i];
  const float inv = 1.0f / tot;
#pragma unroll
  for (int i = 0; i < 8; ++i) attn[(size_t)b * L_ + tid + i * 256] = v[i] * inv;
}

// -------------------------------------------------------------------------
// Kernel 3: wout[b, e] = sum_l attn[b,l] * om[b,l,e]
// A = attention chunk in row M=0 only (zeros elsewhere); B = 4x16 f32 tiles
// of output_memory (K = l rows, N = e columns -> coalesced 64B segments).
// Each wave keeps 4 accumulators (64 e-columns) to amortize the A fragment.
// B 4x16 f32 layout: lanes 0-15 {K=0(v0),K=1(v1)} at N=lane,
//                    lanes 16-31 {K=2(v0),K=3(v1)} at N=lane-16.
// -------------------------------------------------------------------------
__global__ __launch_bounds__(256) void weighted_kernel(
    const float* __restrict__ om, const float* __restrict__ attn,
    float* __restrict__ wout) {
  const int lane = threadIdx.x & 31;
  const int wave = threadIdx.x >> 5;
  const int b    = blockIdx.y;
  const int e0   = (blockIdx.x * 8 + wave) * 64;
  const int n    = lane & 15;
  const int half = lane >> 4;

  const float* __restrict__ arow  = attn + (size_t)b * L_;
  const float* __restrict__ bbase = om + (size_t)b * L_ * E_;

  v8f acc0 = {0.f, 0.f, 0.f, 0.f, 0.f, 0.f, 0.f, 0.f};
  v8f acc1 = acc0, acc2 = acc0, acc3 = acc0;

  for (int l = 0; l < L_; l += 4) {
    // A fragment: row M=0 carries attention weights, all other rows zero.
    const float a0 = arow[l + 2 * half];
    const float a1 = arow[l + 2 * half + 1];
    v2f a;
    a.x = (n == 0) ? a0 : 0.0f;
    a.y = (n == 0) ? a1 : 0.0f;

    const float* p0 = bbase + (size_t)(l + 2 * half) * E_ + e0 + n;
    const float* p1 = p0 + E_;
    v2f b0; b0.x = p0[0];  b0.y = p1[0];
    v2f b1; b1.x = p0[16]; b1.y = p1[16];
    v2f b2; b2.x = p0[32]; b2.y = p1[32];
    v2f b3; b3.x = p0[48]; b3.y = p1[48];

    acc0 = __builtin_amdgcn_wmma_f32_16x16x4_f32(false, a, false, b0, (short)0, acc0, false, false);
    acc1 = __builtin_amdgcn_wmma_f32_16x16x4_f32(false, a, false, b1, (short)0, acc1, false, false);
    acc2 = __builtin_amdgcn_wmma_f32_16x16x4_f32(false, a, false, b2, (short)0, acc2, false, false);
    acc3 = __builtin_amdgcn_wmma_f32_16x16x4_f32(false, a, false, b3, (short)0, acc3, false, false);
  }

  // Row M=0 of D lives in VGPR0, lanes 0-15 (N = lane).
  if (half == 0) {
    float* o = wout + (size_t)b * E_ + e0 + n;
    o[0]  = acc0[0];
    o[16] = acc1[0];
    o[32] = acc2[0];
    o[48] = acc3[0];
  }
}

// -------------------------------------------------------------------------
extern "C" void kernel_launch(void* const* d_in, const int* in_sizes, int n_in,
                              void* d_out, int out_size, void* d_ws, size_t ws_size,
                              hipStream_t stream) {
  (void)in_sizes; (void)n_in; (void)out_size; (void)d_ws; (void)ws_size;
  const float* memory        = (const float*)d_in[0];
  const float* output_memory = (const float*)d_in[1];
  const float* query         = (const float*)d_in[2];
  const int*   memory_mask   = (const int*)d_in[3];
  // d_in[4] = maxlen scalar; compile-time L_ here.

  float* attn = (float*)d_out;                    // [B, L]  (also raw-score scratch)
  float* wout = (float*)d_out + (size_t)B_ * L_;  // [B, E]

  dim3 g1(L_ / 128, B_);
  scores_kernel<<<g1, 256, 0, stream>>>(memory, query, attn);

  softmax_kernel<<<B_, 256, 0, stream>>>(attn, memory_mask);

  dim3 g3(E_ / 512, B_);
  weighted_kernel<<<g3, 256, 0, stream>>>(output_memory, attn, wout);
}